// Brain_50929722196184
// MI455X (gfx1250) — compile-verified
//
#include <hip/hip_runtime.h>
#include <stdint.h>

typedef float v2f __attribute__((ext_vector_type(2)));
typedef float v4f __attribute__((ext_vector_type(4)));
typedef float v8f __attribute__((ext_vector_type(8)));

#define BM 128
#define BN 128
#define BK 16
#define LDSS 20          // padded stride (floats): conflict-free b64 frag reads, 16B alignment
#define TILEF (BM * LDSS)

// ---- CDNA5 async global->LDS copy (bypasses VGPRs, tracked by ASYNCcnt) ----
__device__ __forceinline__ void async_ld_b128(uint32_t lds_addr, const float* g) {
    asm volatile("global_load_async_to_lds_b128 %0, %1, off"
                 :: "v"(lds_addr), "v"((uint64_t)(uintptr_t)g)
                 : "memory");
}
template <int N>
__device__ __forceinline__ void wait_asynccnt() {
    asm volatile("s_wait_asynccnt %0" :: "n"(N) : "memory");
}

// Issue one 128x16 A-tile + one 128x16 B-tile as 4 async b128 copies per thread.
__device__ __forceinline__ void issue_tile(
    const float* __restrict__ A, const float* __restrict__ W, int K, int k,
    int mBase, int nBase, int t, uint32_t asAddr, uint32_t bsAddr)
{
#pragma unroll
    for (int p = 0; p < 2; ++p) {
        const int f   = t + p * 256;          // float4 index 0..511
        const int row = f >> 2;               // 0..127
        const int col = (f & 3) << 2;         // 0,4,8,12
        const uint32_t loff = (uint32_t)((row * LDSS + col) * 4);
        async_ld_b128(asAddr + loff, A + (size_t)(mBase + row) * K + (size_t)(k + col));
        async_ld_b128(bsAddr + loff, W + (size_t)(nBase + row) * K + (size_t)(k + col));
    }
}

// ---------------------------------------------------------------------------
// Fused NT GEMM: out[m,n] = relu?( sum_k A0[m,k]W0[n,k] + sum_k A1[m,k]W1[n,k] + bias[n] )
// fp32 WMMA (V_WMMA_F32_16X16X4_F32) for exact fp32-FMA-class numerics.
// Double-buffered LDS fed by async global->LDS copies.
// ---------------------------------------------------------------------------
template <bool RELU>
__global__ __launch_bounds__(256) void gemm_nt_fused(
    const float* __restrict__ A0, const float* __restrict__ W0, int K0,
    const float* __restrict__ A1, const float* __restrict__ W1, int K1,
    const float* __restrict__ bias, float* __restrict__ out, int N)
{
    __shared__ __align__(16) float As[2][TILEF];
    __shared__ __align__(16) float Bs[2][TILEF];

    const int t     = threadIdx.x;
    const int lane  = t & 31;
    const int l15   = lane & 15;
    const int lhalf = lane >> 4;            // 0|1 -> K-pair select / M+8 select
    const int wid   = t >> 5;               // 8 waves
    const int wm    = (wid >> 2) * 64;      // wave tile 64x32 within 128x128 block
    const int wn    = (wid & 3) * 32;
    const int mBase = blockIdx.y * BM;
    const int nBase = blockIdx.x * BN;

    const uint32_t asAddr[2] = { (uint32_t)(uintptr_t)&As[0][0],
                                 (uint32_t)(uintptr_t)&As[1][0] };
    const uint32_t bsAddr[2] = { (uint32_t)(uintptr_t)&Bs[0][0],
                                 (uint32_t)(uintptr_t)&Bs[1][0] };

    v8f acc[4][2];
#pragma unroll
    for (int mi = 0; mi < 4; ++mi)
#pragma unroll
        for (int ni = 0; ni < 2; ++ni)
            acc[mi][ni] = (v8f)0.0f;

    const int T0 = K0 / BK;
    const int T  = T0 + K1 / BK;

    // Prefetch step 0 into buffer 0
    issue_tile(A0, W0, K0, 0, mBase, nBase, t, asAddr[0], bsAddr[0]);

    int buf = 0;
#pragma unroll 1
    for (int s = 0; s < T; ++s) {
        if (s + 1 < T) {
            const float* A; const float* W; int K; int k;
            if (s + 1 < T0) { A = A0; W = W0; K = K0; k = (s + 1) * BK; }
            else            { A = A1; W = W1; K = K1; k = (s + 1 - T0) * BK; }
            issue_tile(A, W, K, k, mBase, nBase, t, asAddr[buf ^ 1], bsAddr[buf ^ 1]);
            wait_asynccnt<4>();   // in-order: current tile's 4 copies are done
        } else {
            wait_asynccnt<0>();
        }
        __syncthreads();          // all lanes' current-tile data visible

        const float* __restrict__ as = &As[buf][0];
        const float* __restrict__ bs = &Bs[buf][0];
#pragma unroll
        for (int kk = 0; kk < BK; kk += 4) {
            const int fc = kk + (lhalf << 1);   // fp32 A/B frag: K-pair per lane half
            v2f a[4], b[2];
#pragma unroll
            for (int mi = 0; mi < 4; ++mi)
                a[mi] = *(const v2f*)(&as[(wm + mi * 16 + l15) * LDSS + fc]);
#pragma unroll
            for (int ni = 0; ni < 2; ++ni)
                b[ni] = *(const v2f*)(&bs[(wn + ni * 16 + l15) * LDSS + fc]);
#pragma unroll
            for (int mi = 0; mi < 4; ++mi)
#pragma unroll
                for (int ni = 0; ni < 2; ++ni)
                    acc[mi][ni] = __builtin_amdgcn_wmma_f32_16x16x4_f32(
                        false, a[mi], false, b[ni],
                        (short)0, acc[mi][ni], false, false);
        }
        __syncthreads();          // all waves done reading buf -> safe to overwrite
        buf ^= 1;
    }

    // Epilogue: C/D layout -> lane: n = l&15, m = r + lhalf*8 per VGPR r
#pragma unroll
    for (int mi = 0; mi < 4; ++mi) {
#pragma unroll
        for (int ni = 0; ni < 2; ++ni) {
            const int n_g = nBase + wn + ni * 16 + l15;
            const float bv = bias[n_g];
#pragma unroll
            for (int r = 0; r < 8; ++r) {
                const int m_g = mBase + wm + mi * 16 + r + (lhalf << 3);
                float v = acc[mi][ni][r] + bv;
                if (RELU) v = fmaxf(v, 0.0f);
                out[(size_t)m_g * (size_t)N + (size_t)n_g] = v;
            }
        }
    }
}

// ---------------------------------------------------------------------------
// Deterministic two-pass column reductions (no float atomics)
// mode 0: sum |x| ; mode 1: count(|x| > thresh)
// ---------------------------------------------------------------------------
__global__ __launch_bounds__(256) void col_partial(
    const float* __restrict__ src, float* __restrict__ ws,
    int rows_per_chunk, int cols, int mode, float thresh)
{
    const int j = blockIdx.x * 256 + threadIdx.x;
    const int chunk = blockIdx.y;
    const float* p = src + (size_t)chunk * rows_per_chunk * cols + j;
    float s = 0.0f;
#pragma unroll 4
    for (int r = 0; r < rows_per_chunk; ++r) {
        const float v = fabsf(p[(size_t)r * cols]);
        s += (mode == 0) ? v : ((v > thresh) ? 1.0f : 0.0f);
    }
    ws[(size_t)chunk * cols + j] = s;
}

__global__ __launch_bounds__(256) void col_finish(
    const float* __restrict__ ws, float* __restrict__ dst,
    int nchunks, int cols, float scale)
{
    const int j = blockIdx.x * 256 + threadIdx.x;
    float s = 0.0f;
    for (int c = 0; c < nchunks; ++c) s += ws[(size_t)c * cols + j];
    dst[j] = s * scale;
}

// ---------------------------------------------------------------------------
extern "C" void kernel_launch(void* const* d_in, const int* in_sizes, int n_in,
                              void* d_out, int out_size, void* d_ws, size_t ws_size,
                              hipStream_t stream)
{
    (void)in_sizes; (void)n_in; (void)out_size; (void)ws_size;
    constexpr int B = 8192, I = 1024, H = 2048, O = 1024;

    const float* x     = (const float*)d_in[0];
    const float* h     = (const float*)d_in[1];
    const float* W_in  = (const float*)d_in[2];
    const float* W_hh  = (const float*)d_in[3];
    const float* b_h   = (const float*)d_in[4];
    const float* W_out = (const float*)d_in[5];
    const float* b_out = (const float*)d_in[6];

    float* y         = (float*)d_out;                      // [B,O]
    float* h_new     = y + (size_t)B * O;                  // [B,H]
    float* act_stats = h_new + (size_t)B * H;              // [H]
    float* rec_stats = act_stats + H;                      // [H]
    float* ws        = (float*)d_ws;                       // 32*H partials (reused)

    // 1) h_new = relu(x@W_in^T + h@W_hh^T + b_h)   (fused dual-K GEMM)
    gemm_nt_fused<true><<<dim3(H / BN, B / BM), 256, 0, stream>>>(
        x, W_in, I, h, W_hh, H, b_h, h_new, H);

    // 2) act_stats = mean(|h_new|, axis=0)
    constexpr int CH = 32, RPC = B / CH;
    col_partial<<<dim3(H / 256, CH), 256, 0, stream>>>(h_new, ws, RPC, H, 0, 0.0f);
    col_finish<<<H / 256, 256, 0, stream>>>(ws, act_stats, CH, H, 1.0f / (float)B);

    // 3) rec_stats = mean(|h| > 0.1, axis=0)
    col_partial<<<dim3(H / 256, CH), 256, 0, stream>>>(h, ws, RPC, H, 1, 0.1f);
    col_finish<<<H / 256, 256, 0, stream>>>(ws, rec_stats, CH, H, 1.0f / (float)B);

    // 4) y = h_new@W_out^T + b_out
    gemm_nt_fused<false><<<dim3(O / BN, B / BM), 256, 0, stream>>>(
        h_new, W_out, H, nullptr, nullptr, 0, b_out, y, O);
}